// GCN_45887430590687
// MI455X (gfx1250) — compile-verified
//
#include <hip/hip_runtime.h>
#include <hip/hip_bf16.h>

typedef float v2f __attribute__((ext_vector_type(2)));
typedef float v8f __attribute__((ext_vector_type(8)));

#define F_IN  64
#define F_HID 128

// ---------------- CSR build ----------------

__global__ void k_cnt_init(int* __restrict__ cnt, int n) {
    int i = blockIdx.x * 256 + threadIdx.x;
    if (i < n) cnt[i] = 0;
}

__global__ void k_cnt(const long long* __restrict__ dst, int* __restrict__ cnt, int e) {
    int i = blockIdx.x * 256 + threadIdx.x;
    if (i < e) atomicAdd(&cnt[(int)dst[i]], 1);
}

// Single-block exclusive scan over cnt -> roff[0..n]; also zeroes cursors and
// computes dinv[i] = rsqrt(indeg + 1) (self-loop included).
__global__ __launch_bounds__(1024) void k_scan(const int* __restrict__ cnt,
                                               int* __restrict__ roff,
                                               int* __restrict__ cur,
                                               float* __restrict__ dinv, int n) {
    __shared__ int sh[1024];
    __shared__ int carry;
    const int tid = threadIdx.x;
    if (tid == 0) carry = 0;
    __syncthreads();
    const int nch = (n + 1023) >> 10;
    for (int c = 0; c < nch; ++c) {
        int i = (c << 10) + tid;
        int v = (i < n) ? cnt[i] : 0;
        sh[tid] = v;
        __syncthreads();
        for (int off = 1; off < 1024; off <<= 1) {     // Hillis-Steele inclusive scan
            int t = (tid >= off) ? sh[tid - off] : 0;
            __syncthreads();
            sh[tid] += t;
            __syncthreads();
        }
        if (i < n) {
            roff[i] = carry + sh[tid] - v;             // exclusive prefix
            cur[i]  = 0;
            dinv[i] = rsqrtf((float)(v + 1));
        }
        __syncthreads();
        if (tid == 1023) carry += sh[1023];
        __syncthreads();
    }
    if (tid == 0) roff[n] = carry;
}

__global__ void k_fill(const long long* __restrict__ src, const long long* __restrict__ dst,
                       const int* __restrict__ roff, int* __restrict__ cur,
                       int* __restrict__ csr, int e) {
    int i = blockIdx.x * 256 + threadIdx.x;
    if (i < e) {
        int d = (int)dst[i];
        int pos = atomicAdd(&cur[d], 1);
        csr[roff[d] + pos] = (int)src[i];
    }
}

// ---------------- dense transform: C[n,128] = A[n,K] @ W[K,128] ----------------
// One wave32 computes one 16x16 tile with V_WMMA_F32_16X16X4_F32.
// 8 waves per block -> 16 rows x 128 cols per block.

__global__ __launch_bounds__(256) void k_gemm_wmma(
        const float* __restrict__ A, const float* __restrict__ W,
        float* __restrict__ C, int n, int K) {
    const int lane = threadIdx.x & 31;
    const int wave = threadIdx.x >> 5;
    const int half = lane >> 4;               // 0: lanes 0-15, 1: lanes 16-31
    const int l15  = lane & 15;
    const int m0   = blockIdx.x * 16;
    const int n0   = wave * 16;               // 8 waves * 16 = 128 cols

    int arow = m0 + l15;
    if (arow >= n) arow = n - 1;              // keep EXEC all-ones for WMMA

    // A layout (32-bit A 16x4): VGPR0 = K+2*half, VGPR1 = K+2*half+1 -> contiguous pair
    const float* Ap = A + (size_t)arow * K + 2 * half;
    // B layout (32-bit B 4x16): row K striped across lanes within a VGPR
    const float* Wp = W + (size_t)(2 * half) * F_HID + n0 + l15;

    v8f acc = {0.f, 0.f, 0.f, 0.f, 0.f, 0.f, 0.f, 0.f};
    for (int k = 0; k < K; k += 4) {
        v2f a = *(const v2f*)(Ap + k);
        v2f b;
        b.x = Wp[(size_t)k * F_HID];
        b.y = Wp[(size_t)(k + 1) * F_HID];
        acc = __builtin_amdgcn_wmma_f32_16x16x4_f32(
            /*neg_a=*/false, a, /*neg_b=*/false, b,
            /*c_mod=*/(short)0, acc, /*reuse_a=*/false, /*reuse_b=*/false);
    }

    // C/D layout: VGPR r -> row m0 + r + 8*half, col n0 + l15
#pragma unroll
    for (int r = 0; r < 8; ++r) {
        int row = m0 + r + 8 * half;
        if (row < n) C[(size_t)row * F_HID + n0 + l15] = acc[r];
    }
}

// ---------------- fused aggregation: self-loop + CSR gather + bias + ReLU ----------------
// One wave32 per destination node; lane l owns features [4l, 4l+3].
// X[d] = relu( sum_{s in N(d)} dinv[s]*dinv[d]*Y[s] + dinv[d]^2*Y[d] + b )

__global__ __launch_bounds__(256) void k_node_agg(
        const float4* __restrict__ Y, const int* __restrict__ roff,
        const int* __restrict__ csr, const float* __restrict__ dinv,
        const float4* __restrict__ bias, float4* __restrict__ X, int n) {
    const int lane = threadIdx.x & 31;
    const int node = blockIdx.x * 8 + (threadIdx.x >> 5);
    if (node >= n) return;

    const float dd = dinv[node];
    float4 acc = Y[node * 32 + lane];         // self-loop seed
    const float c0 = dd * dd;
    acc.x *= c0; acc.y *= c0; acc.z *= c0; acc.w *= c0;

    const int j0 = roff[node], j1 = roff[node + 1];
    int s = (j0 < j1) ? csr[j0] : 0;
    for (int j = j0; j < j1; ++j) {
        int snext = (j + 1 < j1) ? csr[j + 1] : s;
        __builtin_prefetch(&Y[snext * 32 + lane], 0, 0);   // global_prefetch_b8
        float c = dinv[s] * dd;
        float4 v = Y[s * 32 + lane];
        acc.x += v.x * c; acc.y += v.y * c;
        acc.z += v.z * c; acc.w += v.w * c;
        s = snext;
    }

    float4 bb = bias[lane];                   // 128 bias floats = 32 float4
    acc.x = fmaxf(acc.x + bb.x, 0.f);
    acc.y = fmaxf(acc.y + bb.y, 0.f);
    acc.z = fmaxf(acc.z + bb.z, 0.f);
    acc.w = fmaxf(acc.w + bb.w, 0.f);
    X[node * 32 + lane] = acc;
}

// ---------------- final projection: out[n] = X[n,128] @ Wout + bout ----------------

__global__ __launch_bounds__(256) void k_out(
        const float4* __restrict__ X, const float4* __restrict__ Wout,
        const float* __restrict__ bout, float* __restrict__ out, int n) {
    int lane = threadIdx.x & 31;
    int node = blockIdx.x * 8 + (threadIdx.x >> 5);
    if (node >= n) return;
    float4 h = X[node * 32 + lane];
    float4 w = Wout[lane];
    float s = h.x * w.x + h.y * w.y + h.z * w.z + h.w * w.w;
#pragma unroll
    for (int off = 16; off; off >>= 1) s += __shfl_xor(s, off, 32);
    if (lane == 0) out[node] = s + bout[0];
}

// ---------------- host orchestration ----------------

extern "C" void kernel_launch(void* const* d_in, const int* in_sizes, int n_in,
                              void* d_out, int out_size, void* d_ws, size_t ws_size,
                              hipStream_t stream) {
    const float*     x    = (const float*)d_in[0];
    const long long* ei   = (const long long*)d_in[1];   // int64 [2, E]
    const float*     W0   = (const float*)d_in[2];
    const float*     b0   = (const float*)d_in[3];
    const float*     W1   = (const float*)d_in[4];
    const float*     b1   = (const float*)d_in[5];
    const float*     W2   = (const float*)d_in[6];
    const float*     b2   = (const float*)d_in[7];
    const float*     W3   = (const float*)d_in[8];
    const float*     b3   = (const float*)d_in[9];
    const float*     Wout = (const float*)d_in[10];
    const float*     bout = (const float*)d_in[11];
    float*           outp = (float*)d_out;

    const int N = in_sizes[0] / F_IN;
    const int E = in_sizes[1] / 2;
    const long long* src = ei;
    const long long* dst = ei + E;

    // workspace layout (floats first for float4 alignment, then ints)
    float* ws   = (float*)d_ws;
    float* dinv = ws;                              // [N]
    float* Y    = ws + (size_t)N;                  // [N,128] GEMM output
    float* X    = Y + (size_t)N * F_HID;           // [N,128] post-activation
    int*   ints = (int*)(X + (size_t)N * F_HID);
    int*   cnt  = ints;                            // [N]   in-degree counts
    int*   roff = cnt + N;                         // [N+1] CSR row offsets
    int*   cur  = roff + (N + 1);                  // [N]   fill cursors
    int*   csr  = cur + N;                         // [E]   source indices by dst

    dim3 blk(256);
    int gN    = (N + 255) / 256;
    int gE    = (E + 255) / 256;
    int gGemm = (N + 15) / 16;
    int gNode = (N + 7) / 8;

    // CSR build (once per call)
    k_cnt_init<<<gN, blk, 0, stream>>>(cnt, N);
    k_cnt     <<<gE, blk, 0, stream>>>(dst, cnt, E);
    k_scan    <<<1, dim3(1024), 0, stream>>>(cnt, roff, cur, dinv, N);
    k_fill    <<<gE, blk, 0, stream>>>(src, dst, roff, cur, csr, E);

    const float* Wl[4] = {W0, W1, W2, W3};
    const float* bl[4] = {b0, b1, b2, b3};
    const float* hin = x;
    int K = F_IN;
    for (int layer = 0; layer < 4; ++layer) {
        k_gemm_wmma<<<gGemm, blk, 0, stream>>>(hin, Wl[layer], Y, N, K);
        k_node_agg <<<gNode, blk, 0, stream>>>((const float4*)Y, roff, csr, dinv,
                                               (const float4*)bl[layer], (float4*)X, N);
        hin = X;
        K = F_HID;
    }
    k_out<<<gNode, blk, 0, stream>>>((const float4*)X, (const float4*)Wout, bout, outp, N);
}